// DynamicTaskMemoryInduction_12713103197290
// MI455X (gfx1250) — compile-verified
//
#include <hip/hip_runtime.h>
#include <math.h>

typedef float v2f __attribute__((ext_vector_type(2)));
typedef float v8f __attribute__((ext_vector_type(8)));

#define QROWS 512
#define MROWS 64
#define AR    (QROWS + MROWS)   /* 576 GEMM rows: q rows then m rows */
#define NCAP  4
#define CCAP  64
#define DCAP  192
#define KDIM  768
#define JDIM  (NCAP * DCAP)     /* 768 GEMM cols = (n,d) flattened */
#define EPSF  1e-8f

// ---------------------------------------------------------------------------
// Pack A = [q ; m]  (row-major [576,768])
// ---------------------------------------------------------------------------
__global__ void pack_A_kernel(const float* __restrict__ m, const float* __restrict__ q,
                              float* __restrict__ A) {
  int idx = blockIdx.x * blockDim.x + threadIdx.x;
  if (idx >= AR * KDIM) return;
  int r = idx / KDIM, i = idx - r * KDIM;
  A[idx] = (r < QROWS) ? q[(size_t)r * KDIM + i]
                       : m[(size_t)(r - QROWS) * KDIM + i];
}

// ---------------------------------------------------------------------------
// Pack B[j=n*192+d][i] = W[0][n][0][d][i]  (c=0 slice; W is a broadcast over c,
// so this is bit-identical to every c slice — avoids reading 151 MB of W)
// ---------------------------------------------------------------------------
__global__ void pack_B_kernel(const float* __restrict__ W, float* __restrict__ B) {
  int idx = blockIdx.x * blockDim.x + threadIdx.x;
  if (idx >= JDIM * KDIM) return;
  int j = idx / KDIM, i = idx - j * KDIM;
  int n = j / DCAP, d = j - n * DCAP;
  B[idx] = W[(size_t)(n * CCAP * DCAP + d) * KDIM + i];
}

// ---------------------------------------------------------------------------
// NT GEMM in fp32 via V_WMMA_F32_16X16X4_F32:
//   C[r,j] = sum_i A[r,i] * B[j,i],  M=576, N=768, K=768
// One wave per 16x16 output tile. Fragment layouts per ISA 7.12.2 (32-bit):
//   A(16x4): elem(m,k) -> lane = m + 16*(k/2), vgpr = k%2
//   B(4x16): elem(k,n) -> lane = n + 16*(k/2), vgpr = k%2   (B[k][n] = Bbuf[n][k])
//   C(16x16): elem(m,n) -> lane = n + 16*(m/8), vgpr = m%8
// ---------------------------------------------------------------------------
__global__ __launch_bounds__(256) void gemm_nt_f32_wmma(
    const float* __restrict__ A, const float* __restrict__ B, float* __restrict__ C,
    int M, int N, int K)
{
  const int lane = threadIdx.x & 31;
  const int wave = (int)((blockIdx.x * blockDim.x + threadIdx.x) >> 5);
  const int tilesN = N >> 4;
  const int tm = wave / tilesN;
  const int tn = wave - tm * tilesN;
  if (tm * 16 >= M) return;

  const int r    = tm * 16 + (lane & 15);
  const int cidx = tn * 16 + (lane & 15);
  const int kh   = (lane >> 4) << 1;            // 0 or 2: K-half this lane holds

  const float* ap = A + (size_t)r * K + kh;
  const float* bp = B + (size_t)cidx * K + kh;

  v8f acc = {};
  for (int k = 0; k < K; k += 4) {
    v2f av = *(const v2f*)(ap + k);             // A[r][k+kh], A[r][k+kh+1]
    v2f bv = *(const v2f*)(bp + k);             // B[cidx][k+kh], B[cidx][k+kh+1]
    acc = __builtin_amdgcn_wmma_f32_16x16x4_f32(
        /*neg_a=*/false, av, /*neg_b=*/false, bv,
        /*c_mod=*/(short)0, acc, /*reuse_a=*/false, /*reuse_b=*/false);
  }

  const int mbase = tm * 16 + ((lane >> 4) << 3);
#pragma unroll
  for (int v = 0; v < 8; ++v)
    C[(size_t)(mbase + v) * N + cidx] = acc[v];
}

// ---------------------------------------------------------------------------
// Per-(n,c): tmp_m = hat_m + b, mean over d, centered xm (two layouts), sum xm^2
// ---------------------------------------------------------------------------
__global__ __launch_bounds__(DCAP) void stats_kernel(
    const float* __restrict__ hat, const float* __restrict__ b,
    float* __restrict__ xm_ncd, float* __restrict__ xm_ndc,
    float* __restrict__ meanm, float* __restrict__ sx2)
{
  __shared__ float red[DCAP];
  const int nc = blockIdx.x;                 // 0..255
  const int n = nc / CCAP, c = nc - n * CCAP;
  const int d = threadIdx.x;                 // 0..191

  const float t = hat[(size_t)(QROWS + c) * JDIM + n * DCAP + d] + b[(size_t)nc * DCAP + d];
  red[d] = t;
  __syncthreads();
  for (int s = 96; s > 1; s >>= 1) {         // 192 -> 96 -> ... -> 3
    if (d < s) red[d] += red[d + s];
    __syncthreads();
  }
  const float mean = (red[0] + red[1] + red[2]) * (1.0f / DCAP);
  __syncthreads();

  const float xm = t - mean;
  xm_ncd[(size_t)nc * DCAP + d] = xm;
  xm_ndc[(size_t)(n * DCAP + d) * CCAP + c] = xm;

  red[d] = xm * xm;
  __syncthreads();
  for (int s = 96; s > 1; s >>= 1) {
    if (d < s) red[d] += red[d + s];
    __syncthreads();
  }
  if (d == 0) {
    meanm[nc] = mean;
    sx2[nc]   = red[0] + red[1] + red[2];
  }
}

// ---------------------------------------------------------------------------
// Routing: one block per query. tmp_q is c-independent throughout, so routing
// state per q is just [4][192] vectors + [4][64] scalars in LDS.
// ---------------------------------------------------------------------------
__global__ __launch_bounds__(256) void routing_kernel(
    const float* __restrict__ hat,
    const float* __restrict__ xm_ncd,
    const float* __restrict__ xm_ndc,
    const float* __restrict__ meanm,
    const float* __restrict__ sx2,
    float* __restrict__ out)
{
  __shared__ float tq[NCAP][DCAP];
  __shared__ float ym[NCAP][DCAP];
  __shared__ float hv[NCAP][DCAP];
  __shared__ float vv[NCAP][DCAP];
  __shared__ float p_s[NCAP][CCAP];
  __shared__ float a_s[NCAP][CCAP];
  __shared__ float coef[NCAP][CCAP];
  __shared__ float mm_s[NCAP][CCAP];
  __shared__ float meanq_s[NCAP];
  __shared__ float sy2_s[NCAP];
  __shared__ float sq_s[NCAP];

  const int qi   = blockIdx.x;
  const int tid  = threadIdx.x;
  const int lane = tid & 31;
  const int wv   = tid >> 5;        // wave id 0..7 (wave32)
  const int nn   = tid / CCAP;      // 0..3
  const int cc   = tid - nn * CCAP; // 0..63

  for (int idx = tid; idx < NCAP * DCAP; idx += 256)
    tq[idx / DCAP][idx % DCAP] = hat[(size_t)qi * JDIM + idx];
  a_s[nn][cc]  = 0.0f;
  mm_s[nn][cc] = meanm[tid];
  __syncthreads();

  auto waveRed = [&](float v) {
#pragma unroll
    for (int off = 16; off > 0; off >>= 1)
      v += __shfl_xor(v, off, 32);
    return v;
  };

  // center tmp_q over d, compute sy2, then p = tanh(-pearson)
  auto centerP = [&]() {
    if (wv < NCAP) {
      float s = 0.0f;
      for (int d = lane; d < DCAP; d += 32) s += tq[wv][d];
      s = waveRed(s);
      if (lane == 0) meanq_s[wv] = s * (1.0f / DCAP);
    }
    __syncthreads();
    for (int idx = tid; idx < NCAP * DCAP; idx += 256) {
      int n = idx / DCAP, d = idx % DCAP;
      ym[n][d] = tq[n][d] - meanq_s[n];
    }
    __syncthreads();
    if (wv < NCAP) {
      float s = 0.0f;
      for (int d = lane; d < DCAP; d += 32) { float t = ym[wv][d]; s += t * t; }
      s = waveRed(s);
      if (lane == 0) sy2_s[wv] = s;
    }
    __syncthreads();
    {
      const float* xr = xm_ndc + (size_t)nn * DCAP * CCAP + cc;  // coalesced in cc
      float num = 0.0f;
      for (int d = 0; d < DCAP; ++d) num += xr[(size_t)d * CCAP] * ym[nn][d];
      float den = sqrtf(sx2[tid] * sy2_s[nn] + EPSF);
      p_s[nn][cc] = tanhf(-num / den);
    }
    __syncthreads();
  };

  // coef = softmax_n(a) + p ; hv = sum_c coef*tmp_m ; vv = squash(hv)
  auto softHatSquash = [&]() {
    {
      float a0 = a_s[0][cc], a1 = a_s[1][cc], a2 = a_s[2][cc], a3 = a_s[3][cc];
      float mx = fmaxf(fmaxf(a0, a1), fmaxf(a2, a3));
      float e0 = expf(a0 - mx), e1 = expf(a1 - mx), e2 = expf(a2 - mx), e3 = expf(a3 - mx);
      float inv = 1.0f / (e0 + e1 + e2 + e3);
      float en = (nn == 0) ? e0 : (nn == 1) ? e1 : (nn == 2) ? e2 : e3;
      coef[nn][cc] = en * inv + p_s[nn][cc];
    }
    __syncthreads();
    for (int idx = tid; idx < NCAP * DCAP; idx += 256) {
      int n = idx / DCAP, d = idx % DCAP;
      const float* xr = xm_ncd + (size_t)n * CCAP * DCAP + d;    // coalesced in d
      float s = 0.0f;
      for (int c = 0; c < CCAP; ++c)
        s += coef[n][c] * (xr[(size_t)c * DCAP] + mm_s[n][c]);
      hv[n][d] = s;
    }
    __syncthreads();
    if (wv < NCAP) {
      float s = 0.0f;
      for (int d = lane; d < DCAP; d += 32) { float t = hv[wv][d]; s += t * t; }
      s = waveRed(s);
      if (lane == 0) sq_s[wv] = s;
    }
    __syncthreads();
    for (int idx = tid; idx < NCAP * DCAP; idx += 256) {
      int n = idx / DCAP, d = idx % DCAP;
      float sq = sq_s[n];
      float scale = (sq / (1.0f + sq)) / sqrtf(sq + EPSF);
      vv[n][d] = hv[n][d] * scale;
    }
    __syncthreads();
  };

  centerP();                                  // p from original tmp_q
  for (int it = 0; it < 3; ++it) {            // NUM_ROUTING = 3
    softHatSquash();
    if (it == 2) break;                       // final round: no state update
    {
      // agree = sum_d tmp_m * v ; a += p * agree
      const float* xr = xm_ndc + (size_t)nn * DCAP * CCAP + cc;
      float mm = mm_s[nn][cc];
      float s = 0.0f;
      for (int d = 0; d < DCAP; ++d) s += (xr[(size_t)d * CCAP] + mm) * vv[nn][d];
      a_s[nn][cc] += p_s[nn][cc] * s;
    }
    __syncthreads();
    for (int idx = tid; idx < NCAP * DCAP; idx += 256) {
      int n = idx / DCAP, d = idx % DCAP;
      tq[n][d] = (tq[n][d] + vv[n][d]) * 0.5f;
    }
    __syncthreads();
    centerP();
  }

  for (int idx = tid; idx < NCAP * DCAP; idx += 256)
    out[(size_t)qi * JDIM + idx] = vv[idx / DCAP][idx % DCAP];
}

// ---------------------------------------------------------------------------
extern "C" void kernel_launch(void* const* d_in, const int* in_sizes, int n_in,
                              void* d_out, int out_size, void* d_ws, size_t ws_size,
                              hipStream_t stream) {
  const float* m = (const float*)d_in[0];   // [64,768]
  const float* q = (const float*)d_in[1];   // [512,768]
  const float* W = (const float*)d_in[2];   // [1,4,64,192,768] (broadcast over c)
  const float* b = (const float*)d_in[3];   // [1,4,64,192]
  float* out = (float*)d_out;               // [512,768]

  float* ws      = (float*)d_ws;
  float* Abuf    = ws;                                  // 576*768
  float* Bbuf    = Abuf + (size_t)AR * KDIM;            // 768*768
  float* hat     = Bbuf + (size_t)JDIM * KDIM;          // 576*768 (q-rows then m-rows)
  float* xm_ncd  = hat + (size_t)AR * JDIM;             // 4*64*192
  float* xm_ndc  = xm_ncd + (size_t)NCAP * CCAP * DCAP; // 4*192*64
  float* meanmv  = xm_ndc + (size_t)NCAP * CCAP * DCAP; // 256
  float* sx2v    = meanmv + NCAP * CCAP;                // 256
  // total ~6.3 MB of workspace

  pack_A_kernel<<<(AR * KDIM + 255) / 256, 256, 0, stream>>>(m, q, Abuf);
  pack_B_kernel<<<(JDIM * KDIM + 255) / 256, 256, 0, stream>>>(W, Bbuf);

  const int tiles = (AR / 16) * (JDIM / 16);            // 36 * 48 = 1728 waves
  gemm_nt_f32_wmma<<<tiles * 32 / 256, 256, 0, stream>>>(Abuf, Bbuf, hat, AR, JDIM, KDIM);

  stats_kernel<<<NCAP * CCAP, DCAP, 0, stream>>>(hat, b, xm_ncd, xm_ndc, meanmv, sx2v);
  routing_kernel<<<QROWS, 256, 0, stream>>>(hat, xm_ncd, xm_ndc, meanmv, sx2v, out);
}